// LSTMBrain_17789754540387
// MI455X (gfx1250) — compile-verified
//
#include <hip/hip_runtime.h>

// CDNA5 / gfx1250: wave32, WMMA 16x16x4 f32.
typedef float v2f __attribute__((ext_vector_type(2)));
typedef float v8f __attribute__((ext_vector_type(8)));

#define GATES 12  // 4 * HID, HID = 3

__device__ __forceinline__ float fast_rcp(float x) {
#if __has_builtin(__builtin_amdgcn_rcpf)
  return __builtin_amdgcn_rcpf(x);
#else
  return 1.0f / x;
#endif
}

__device__ __forceinline__ float fast_exp2(float x) {
#if __has_builtin(__builtin_amdgcn_exp2f)
  return __builtin_amdgcn_exp2f(x);
#else
  return exp2f(x);
#endif
}

// sigmoid(x) = 1 / (1 + 2^(-x*log2e)); safe at both extremes.
__device__ __forceinline__ float fsig(float x) {
  return fast_rcp(1.0f + fast_exp2(-1.4426950408889634f * x));
}

// tanh(x) = (1 - e^-2|x|)/(1 + e^-2|x|) with sign restore; avoids inf*0 NaN.
__device__ __forceinline__ float ftanh(float x) {
  float ax = __builtin_fabsf(x);
  float e = fast_exp2(-2.8853900817779268f * ax);
  float r = (1.0f - e) * fast_rcp(1.0f + e);
  return __builtin_copysignf(r, x);
}

// One wave handles 16 batch rows. Gate GEMMs per step run on the matrix pipe:
//   layer1: Z = [U1^T | W1^T](16x4) x [h1 ; x_t](4x16) + b1   (1 WMMA)
//   layer2: Z = U2^T x h2 + b2 ; Z += W2^T x h1_new           (2 WMMAs)
__global__ void __launch_bounds__(256)
lstm2_wmma_kernel(const float* __restrict__ x,   // [B, T]
                  const float* __restrict__ W1,  // [1, 12]
                  const float* __restrict__ U1,  // [3, 12]
                  const float* __restrict__ b1,  // [12]
                  const float* __restrict__ W2,  // [3, 12]
                  const float* __restrict__ U2,  // [3, 12]
                  const float* __restrict__ b2,  // [12]
                  const float* __restrict__ Wd,  // [3, 1]
                  const float* __restrict__ bd,  // [1]
                  float* __restrict__ out,       // [B] (reference output [1,B])
                  int B, int T) {
  const int lane = threadIdx.x & 31;
  const int m    = lane & 15;          // batch slot (N) and A-matrix row (M=gate)
  const bool hi  = lane >= 16;
  const int wave = blockIdx.x * (blockDim.x >> 5) + (threadIdx.x >> 5);
  const int batchBase = wave * 16;
  if (batchBase >= B) return;          // uniform per wave

  const int row  = batchBase + m;
  const int rowc = row < B ? row : (B - 1);   // clamp for partial tiles

  // ---- constant A operands (16x4 f32 A layout: lanes 0-15 hold K=0,1;
  //      lanes 16-31 hold K=2,3; row M = lane&15) ----
  float a1x = 0.f, a1y = 0.f, au2x = 0.f, au2y = 0.f, aw2x = 0.f, aw2y = 0.f;
  if (m < GATES) {
    if (!hi) {
      a1x  = U1[0 * GATES + m];  a1y  = U1[1 * GATES + m];
      au2x = U2[0 * GATES + m];  au2y = U2[1 * GATES + m];
      aw2x = W2[0 * GATES + m];  aw2y = W2[1 * GATES + m];
    } else {
      a1x  = U1[2 * GATES + m];  a1y  = W1[m];   // K=3 column carries W1 (input term)
      au2x = U2[2 * GATES + m];  au2y = 0.f;     // K=3 column zero
      aw2x = W2[2 * GATES + m];  aw2y = 0.f;
    }
  }
  v2f A1;  A1.x  = a1x;  A1.y  = a1y;
  v2f AU2; AU2.x = au2x; AU2.y = au2y;
  v2f AW2; AW2.x = aw2x; AW2.y = aw2y;

  // ---- bias as C operand (16x16 f32 C/D layout: VGPR r -> M=r (lanes<16) / r+8) ----
  v8f CB1, CB2;
#pragma unroll
  for (int r = 0; r < 8; ++r) {
    int M = r + (hi ? 8 : 0);
    CB1[r] = (M < GATES) ? b1[M] : 0.0f;
    CB2[r] = (M < GATES) ? b2[M] : 0.0f;
  }

  // recurrent state, valid in lanes 0-15 (one batch element per lane)
  float h1_0 = 0.f, h1_1 = 0.f, h1_2 = 0.f, c1_0 = 0.f, c1_1 = 0.f, c1_2 = 0.f;
  float h2_0 = 0.f, h2_1 = 0.f, h2_2 = 0.f, c2_0 = 0.f, c2_1 = 0.f, c2_2 = 0.f;

  const float* xrow = x + (size_t)rowc * (size_t)T;

#pragma unroll 1
  for (int t = 0; t < T; t += 4) {
    float4 xb = *reinterpret_cast<const float4*>(xrow + t);
#pragma unroll
    for (int k = 0; k < 4; ++k) {
      const float xt = (k == 0) ? xb.x : (k == 1) ? xb.y : (k == 2) ? xb.z : xb.w;

      // ================= layer 1 =================
      // B operand (4x16 f32: lanes 0-15 hold K=0,1; lanes 16-31 hold K=2,3)
      float h1_2s = __shfl_xor(h1_2, 16, 32);
      v2f Bm1; Bm1.x = hi ? h1_2s : h1_0; Bm1.y = hi ? xt : h1_1;
      v8f d = __builtin_amdgcn_wmma_f32_16x16x4_f32(false, A1, false, Bm1,
                                                    (short)0, CB1, false, false);
      // gates 8..11 live in the partner (upper) lane's V0..V3
      float z8  = __shfl_xor(d[0], 16, 32);
      float z9  = __shfl_xor(d[1], 16, 32);
      float z10 = __shfl_xor(d[2], 16, 32);
      float z11 = __shfl_xor(d[3], 16, 32);
      {
        float i0 = fsig(d[0]), i1 = fsig(d[1]), i2 = fsig(d[2]);
        float f0 = fsig(d[3]), f1 = fsig(d[4]), f2 = fsig(d[5]);
        float g0 = ftanh(d[6]), g1 = ftanh(d[7]), g2 = ftanh(z8);
        float o0 = fsig(z9), o1 = fsig(z10), o2 = fsig(z11);
        c1_0 = f0 * c1_0 + i0 * g0;
        c1_1 = f1 * c1_1 + i1 * g1;
        c1_2 = f2 * c1_2 + i2 * g2;
        h1_0 = o0 * ftanh(c1_0);
        h1_1 = o1 * ftanh(c1_1);
        h1_2 = o2 * ftanh(c1_2);
      }

      // ================= layer 2 =================
      float h1n2s = __shfl_xor(h1_2, 16, 32);
      v2f Bh1; Bh1.x = hi ? h1n2s : h1_0; Bh1.y = hi ? 0.0f : h1_1;
      float h2_2s = __shfl_xor(h2_2, 16, 32);
      v2f Bh2; Bh2.x = hi ? h2_2s : h2_0; Bh2.y = hi ? 0.0f : h2_1;

      v8f d2 = __builtin_amdgcn_wmma_f32_16x16x4_f32(false, AU2, false, Bh2,
                                                     (short)0, CB2, false, false);
      d2 = __builtin_amdgcn_wmma_f32_16x16x4_f32(false, AW2, false, Bh1,
                                                 (short)0, d2, false, false);
      float y8  = __shfl_xor(d2[0], 16, 32);
      float y9  = __shfl_xor(d2[1], 16, 32);
      float y10 = __shfl_xor(d2[2], 16, 32);
      float y11 = __shfl_xor(d2[3], 16, 32);
      {
        float i0 = fsig(d2[0]), i1 = fsig(d2[1]), i2 = fsig(d2[2]);
        float f0 = fsig(d2[3]), f1 = fsig(d2[4]), f2 = fsig(d2[5]);
        float g0 = ftanh(d2[6]), g1 = ftanh(d2[7]), g2 = ftanh(y8);
        float o0 = fsig(y9), o1 = fsig(y10), o2 = fsig(y11);
        c2_0 = f0 * c2_0 + i0 * g0;
        c2_1 = f1 * c2_1 + i1 * g1;
        c2_2 = f2 * c2_2 + i2 * g2;
        h2_0 = o0 * ftanh(c2_0);
        h2_1 = o1 * ftanh(c2_1);
        h2_2 = o2 * ftanh(c2_2);
      }
    }
  }

  // dense head: out[b] = h2 . Wd + bd  (valid state lives in lanes 0-15)
  if (!hi && row < B) {
    out[row] = h2_0 * Wd[0] + h2_1 * Wd[1] + h2_2 * Wd[2] + bd[0];
  }
}

extern "C" void kernel_launch(void* const* d_in, const int* in_sizes, int n_in,
                              void* d_out, int out_size, void* d_ws, size_t ws_size,
                              hipStream_t stream) {
  const float* x  = (const float*)d_in[0];
  const float* W1 = (const float*)d_in[1];
  const float* U1 = (const float*)d_in[2];
  const float* b1 = (const float*)d_in[3];
  const float* W2 = (const float*)d_in[4];
  const float* U2 = (const float*)d_in[5];
  const float* b2 = (const float*)d_in[6];
  const float* Wd = (const float*)d_in[7];
  const float* bd = (const float*)d_in[8];
  float* out = (float*)d_out;

  const int B = out_size;               // output is [1, B]
  const int T = in_sizes[0] / B;        // state is [B, T]

  const int wavesTotal    = (B + 15) / 16;   // 16 sequences per wave
  const int wavesPerBlock = 8;               // 256 threads/block
  const int blocks = (wavesTotal + wavesPerBlock - 1) / wavesPerBlock;

  lstm2_wmma_kernel<<<blocks, wavesPerBlock * 32, 0, stream>>>(
      x, W1, U1, b1, W2, U2, b2, Wd, bd, out, B, T);
}